// TTEmbedding_80994493268368
// MI455X (gfx1250) — compile-verified
//
#include <hip/hip_runtime.h>

typedef float v2f __attribute__((ext_vector_type(2)));
typedef float v8f __attribute__((ext_vector_type(8)));

#define NTOK  16384   // 8 * 2048 tokens
#define DIM   1024
#define RANK  16

// ---------------------------------------------------------------------------
// Kernel 1: per-token left chain  L = core0[i0,:] @ core1[:,i1,:] @ core2[:,i2,:]
// Writes A row (32 floats): L placed at [parity*16 .. parity*16+15], zeros else.
// core0: (1,32,16)  core1: (16,32,16)  core2: (16,32,16), all f32 row-major.
// ---------------------------------------------------------------------------
__global__ void __launch_bounds__(256)
tt_prep_left(const float* __restrict__ core0,
             const float* __restrict__ core1,
             const float* __restrict__ core2,
             const int*   __restrict__ indices,
             float* __restrict__ A) {
  int t = blockIdx.x * blockDim.x + threadIdx.x;
  if (t >= NTOK) return;
  unsigned v = (unsigned)indices[t] & 0xFFFFu;   // vocab index
  unsigned w = v >> 1;
  unsigned i0 = (w >> 10) & 31u;
  unsigned i1 = (w >> 5)  & 31u;
  unsigned i2 =  w        & 31u;
  unsigned parity = v & 1u;

  float L0[RANK], L1[RANK];
#pragma unroll
  for (int r = 0; r < RANK; ++r) L0[r] = core0[i0 * RANK + r];

#pragma unroll
  for (int r = 0; r < RANK; ++r) {
    float acc = 0.f;
#pragma unroll
    for (int p = 0; p < RANK; ++p)
      acc += L0[p] * core1[p * (32 * RANK) + i1 * RANK + r];
    L1[r] = acc;
  }

  float* Arow = A + (size_t)t * 32;
#pragma unroll
  for (int r = 0; r < RANK; ++r) {
    float acc = 0.f;
#pragma unroll
    for (int p = 0; p < RANK; ++p)
      acc += L1[p] * core2[p * (32 * RANK) + i2 * RANK + r];
    Arow[parity * 16 + r]        = acc;
    Arow[(1u - parity) * 16 + r] = 0.f;
  }
}

// ---------------------------------------------------------------------------
// Kernel 2: B (32 x 1024) where B[h*16 + r][d] = sum_s core3[r, t>>6, s] *
// core4[s, t&63],  t = h*1024 + d.   core3: (16,32,16), core4: (16,64,1).
// ---------------------------------------------------------------------------
__global__ void __launch_bounds__(256)
tt_prep_right(const float* __restrict__ core3,
              const float* __restrict__ core4,
              float* __restrict__ B) {
  int tid = blockIdx.x * blockDim.x + threadIdx.x;   // 0 .. 32*DIM-1
  if (tid >= 32 * DIM) return;
  int rp = tid >> 10;          // 0..31  (row of B)
  int d  = tid & (DIM - 1);    // 0..1023
  int h  = rp >> 4;            // parity half
  int r  = rp & 15;            // rank index
  int tt = h * 1024 + d;       // 0..2047
  int i3 = tt >> 6;            // 0..31
  int i4 = tt & 63;            // 0..63
  float acc = 0.f;
#pragma unroll
  for (int s = 0; s < RANK; ++s)
    acc += core3[r * (32 * RANK) + i3 * RANK + s] * core4[s * 64 + i4];
  B[rp * DIM + d] = acc;
}

// ---------------------------------------------------------------------------
// Kernel 3: OUT (16384x1024) = A (16384x32) @ B (32x1024) with
// V_WMMA_F32_16X16X4_F32 (f32 precision).  One wave32 per 16x16 tile,
// 8 chained K=4 WMMAs.  Pure HBM-store-bound (~64 MB out).
//
// A layout (16x4 f32): lanes 0-15: vgpr0=K0, vgpr1=K1 ; lanes 16-31: K2,K3.
// B layout (4x16 f32): mirrored, N = lane&15.
// C/D layout: vgpr j -> row (j + 8*(lane>=16)), col (lane&15).
// ---------------------------------------------------------------------------
__global__ void __launch_bounds__(256)
tt_gemm_wmma(const float* __restrict__ A,   // NTOK x 32
             const float* __restrict__ B,   // 32 x DIM
             float* __restrict__ out) {     // NTOK x DIM
  const int lane      = threadIdx.x & 31;
  const int waveInBlk = threadIdx.x >> 5;
  const int wave      = blockIdx.x * 8 + waveInBlk;  // 65536 waves total
  const int ttile = wave >> 6;    // 0..1023 token tiles
  const int dtile = wave & 63;    // 0..63   dim tiles
  const int halfSel = lane >> 4;  // 0 or 1
  const int l15     = lane & 15;

  const float* Abase = A + (size_t)(ttile * 16 + l15) * 32 + halfSel * 2;
  const float* Bbase = B + dtile * 16 + l15;

  v8f c = {0.f, 0.f, 0.f, 0.f, 0.f, 0.f, 0.f, 0.f};
#pragma unroll
  for (int kc = 0; kc < 8; ++kc) {
    v2f a = *(const v2f*)(Abase + kc * 4);           // K = kc*4 + 2*halfSel {+0,+1}
    v2f b;
    const int k0 = kc * 4 + halfSel * 2;
    b.x = Bbase[(size_t)(k0 + 0) * DIM];
    b.y = Bbase[(size_t)(k0 + 1) * DIM];
    // (neg_a, A, neg_b, B, c_mod, C, reuse_a, reuse_b)
    c = __builtin_amdgcn_wmma_f32_16x16x4_f32(false, a, false, b,
                                              (short)0, c, false, false);
  }

  float* obase = out + (size_t)(ttile * 16 + 8 * halfSel) * DIM + dtile * 16 + l15;
#pragma unroll
  for (int j = 0; j < 8; ++j)
    __builtin_nontemporal_store(c[j], obase + (size_t)j * DIM);
}

// ---------------------------------------------------------------------------
extern "C" void kernel_launch(void* const* d_in, const int* in_sizes, int n_in,
                              void* d_out, int out_size, void* d_ws, size_t ws_size,
                              hipStream_t stream) {
  const float* core0   = (const float*)d_in[0];
  const float* core1   = (const float*)d_in[1];
  const float* core2   = (const float*)d_in[2];
  const float* core3   = (const float*)d_in[3];
  const float* core4   = (const float*)d_in[4];
  const int*   indices = (const int*)d_in[5];
  float* out = (float*)d_out;

  float* A = (float*)d_ws;                 // NTOK*32 f32 = 2 MB
  float* B = A + (size_t)NTOK * 32;        // 32*DIM f32  = 128 KB

  tt_prep_left <<<NTOK / 256, 256, 0, stream>>>(core0, core1, core2, indices, A);
  tt_prep_right<<<(32 * DIM) / 256, 256, 0, stream>>>(core3, core4, B);

  const int totalWaves = (NTOK / 16) * (DIM / 16);   // 65536
  tt_gemm_wmma <<<totalWaves / 8, 256, 0, stream>>>(A, B, out);
}